// OptNet_42365557408283
// MI455X (gfx1250) — compile-verified
//
#include <hip/hip_runtime.h>
#include <hip/hip_bf16.h>

// ---------------------------------------------------------------------------
// OptNet batched KKT solve via Schur complement (Q = q*I):
//   S = A A^T;  S lam = A*puzz - q*b;  z = (puzz - A^T lam)/q
// Hot loop: S = A A^T on V_WMMA_F32_16X16X4_F32, 32x32 register block per
// wave (4 accumulators), branch-free inner loop (clamped pointers, peeled
// K tail). Working set (~90 MB) fits in MI455X's 192 MB L2.
// ---------------------------------------------------------------------------

#define BATCH 64
#define NX    729          // 9*9*9
#define NEQ   324
#define LD    336          // 21*16, padded leading dim for S
#define NTB   11           // ceil(324/32) 32-row block-tiles

typedef __attribute__((ext_vector_type(2))) float v2f;
typedef __attribute__((ext_vector_type(8))) float v8f;

static __device__ __forceinline__ v8f wmma_f32(v2f a, v2f b, v8f c) {
    return __builtin_amdgcn_wmma_f32_16x16x4_f32(
        /*neg_a=*/false, a, /*neg_b=*/false, b,
        /*c_mod=*/(short)0, c, /*reuse_a=*/false, /*reuse_b=*/false);
}

// ---------------------------------------------------------------------------
// 1) r[b,i] = dot(A[b,i,:], puzz[b,:]) - q * bvec[b,i]
// ---------------------------------------------------------------------------
__global__ __launch_bounds__(128)
void optnet_rhs(const float* __restrict__ puzz, const float* __restrict__ A,
                const float* __restrict__ bvec, const float* __restrict__ Q,
                float* __restrict__ r) {
    const int b = blockIdx.y;
    const int i = blockIdx.x * blockDim.x + threadIdx.x;
    if (i >= NEQ) return;
    const float q = Q[0];                       // Q = q*I
    const float* __restrict__ arow = A + (size_t)b * NEQ * NX + (size_t)i * NX;
    const float* __restrict__ pz   = puzz + (size_t)b * NX;
    float acc = 0.0f;
    for (int k = 0; k < NX; ++k) acc = fmaf(arow[k], pz[k], acc);
    r[b * LD + i] = acc - q * bvec[b * NEQ + i];
}

// ---------------------------------------------------------------------------
// 2) S = A A^T, lower triangle, one wave32 per 32x32 block (2x2 WMMA tiles).
//    Out-of-range rows use clamped (row 0) pointers: their products only
//    pollute D rows/cols that are never stored, so the loop is branch-free.
//    K tail (k=728) is peeled into one masked epilogue WMMA.
// ---------------------------------------------------------------------------
__global__ __launch_bounds__(32)
void optnet_s_tiles(const float* __restrict__ A, float* __restrict__ S) {
    const int bi = blockIdx.x;
    const int bj = blockIdx.y;
    if (bj > bi) return;                        // lower triangle (uniform exit)
    const int b  = blockIdx.z;

    const float* __restrict__ Ab = A + (size_t)b * NEQ * NX;

    const int lane = threadIdx.x & 31;          // wave32
    const int m    = lane & 15;
    const int h    = lane >> 4;                 // half-wave selects K pair

    const int ri0 = bi * 32 + m,  ri1 = ri0 + 16;
    const int rj0 = bj * 32 + m,  rj1 = rj0 + 16;
    const float* __restrict__ pi0 = Ab + (size_t)(ri0 < NEQ ? ri0 : 0) * NX;
    const float* __restrict__ pi1 = Ab + (size_t)(ri1 < NEQ ? ri1 : 0) * NX;
    const float* __restrict__ pj0 = Ab + (size_t)(rj0 < NEQ ? rj0 : 0) * NX;
    const float* __restrict__ pj1 = Ab + (size_t)(rj1 < NEQ ? rj1 : 0) * NX;

    v8f c00 = {}, c01 = {}, c10 = {}, c11 = {};
    const int kb = 2 * h;                       // A/B f32 layout: lanes16-31 -> K=2,3

    // main loop: K = 0..727, unconditionally in-bounds (max index 724+3=727)
    #pragma unroll 2
    for (int k0 = 0; k0 <= 724; k0 += 4) {
        const int k = k0 + kb;
        v2f a0, a1, b0, b1;
        a0.x = pi0[k]; a0.y = pi0[k + 1];
        a1.x = pi1[k]; a1.y = pi1[k + 1];
        b0.x = pj0[k]; b0.y = pj0[k + 1];
        b1.x = pj1[k]; b1.y = pj1[k + 1];
        c00 = wmma_f32(a0, b0, c00);
        c01 = wmma_f32(a0, b1, c01);
        c10 = wmma_f32(a1, b0, c10);
        c11 = wmma_f32(a1, b1, c11);
    }

    // peeled tail: K = 728 only (h==1 half-wave and .y components are zero)
    {
        const float ta0 = pi0[728], ta1 = pi1[728];
        const float tb0 = pj0[728], tb1 = pj1[728];
        v2f a0 = { h ? 0.0f : ta0, 0.0f };
        v2f a1 = { h ? 0.0f : ta1, 0.0f };
        v2f b0 = { h ? 0.0f : tb0, 0.0f };
        v2f b1 = { h ? 0.0f : tb1, 0.0f };
        c00 = wmma_f32(a0, b0, c00);
        c01 = wmma_f32(a0, b1, c01);
        c10 = wmma_f32(a1, b0, c10);
        c11 = wmma_f32(a1, b1, c11);
    }

    // D layout: VGPR v, lanes0-15 -> M=v, lanes16-31 -> M=v+8; N = lane&15.
    float* __restrict__ Sb = S + (size_t)b * LD * LD;
    const int col0 = bj * 32 + m, col1 = col0 + 16;
    #pragma unroll
    for (int v = 0; v < 8; ++v) {
        const int row0 = bi * 32 + v + 8 * h;
        const int row1 = row0 + 16;
        if (row0 < NEQ && col0 < NEQ) Sb[row0 * LD + col0] = c00[v];
        if (row0 < NEQ && col1 < NEQ) Sb[row0 * LD + col1] = c01[v];
        if (row1 < NEQ && col0 < NEQ) Sb[row1 * LD + col0] = c10[v];
        if (row1 < NEQ && col1 < NEQ) Sb[row1 * LD + col1] = c11[v];
    }
}

// ---------------------------------------------------------------------------
// 3) In-place right-looking Cholesky of S (lower), one block per batch.
// ---------------------------------------------------------------------------
__global__ __launch_bounds__(256)
void optnet_chol(float* __restrict__ S) {
    float* __restrict__ M = S + (size_t)blockIdx.x * LD * LD;
    const int tid = threadIdx.x;
    for (int j = 0; j < NEQ; ++j) {
        if (tid == 0) M[j * LD + j] = sqrtf(M[j * LD + j]);
        __syncthreads();
        const float inv = 1.0f / M[j * LD + j];
        for (int i = j + 1 + tid; i < NEQ; i += blockDim.x) M[i * LD + j] *= inv;
        __syncthreads();
        for (int i = j + 1 + tid; i < NEQ; i += blockDim.x) {
            const float lij = M[i * LD + j];
            for (int cjj = j + 1; cjj <= i; ++cjj)
                M[i * LD + cjj] = fmaf(-lij, M[cjj * LD + j], M[i * LD + cjj]);
        }
        __syncthreads();
    }
}

// ---------------------------------------------------------------------------
// 4) Solve L L^T lam = r in place (r becomes lam), one block per batch.
// ---------------------------------------------------------------------------
__global__ __launch_bounds__(256)
void optnet_solve(const float* __restrict__ S, float* __restrict__ r) {
    const float* __restrict__ M = S + (size_t)blockIdx.x * LD * LD;
    float* __restrict__ x = r + (size_t)blockIdx.x * LD;
    const int tid = threadIdx.x;
    for (int j = 0; j < NEQ; ++j) {             // forward: L y = r
        if (tid == 0) x[j] = x[j] / M[j * LD + j];
        __syncthreads();
        const float yj = x[j];
        for (int i = j + 1 + tid; i < NEQ; i += blockDim.x)
            x[i] = fmaf(-M[i * LD + j], yj, x[i]);
        __syncthreads();
    }
    for (int j = NEQ - 1; j >= 0; --j) {        // backward: L^T lam = y
        if (tid == 0) x[j] = x[j] / M[j * LD + j];
        __syncthreads();
        const float xj = x[j];
        for (int i = tid; i < j; i += blockDim.x)
            x[i] = fmaf(-M[j * LD + i], xj, x[i]);
        __syncthreads();
    }
}

// ---------------------------------------------------------------------------
// 5) z[b,k] = (puzz[b,k] - sum_i A[b,i,k]*lam[b,i]) / q   (coalesced over k)
// ---------------------------------------------------------------------------
__global__ __launch_bounds__(256)
void optnet_recover(const float* __restrict__ puzz, const float* __restrict__ A,
                    const float* __restrict__ Q, const float* __restrict__ lam,
                    float* __restrict__ out) {
    const int b = blockIdx.y;
    const int k = blockIdx.x * blockDim.x + threadIdx.x;
    if (k >= NX) return;
    const float qinv = 1.0f / Q[0];
    const float* __restrict__ Ab = A + (size_t)b * NEQ * NX;
    const float* __restrict__ lb = lam + (size_t)b * LD;
    float acc = puzz[(size_t)b * NX + k];
    for (int i = 0; i < NEQ; ++i) acc = fmaf(-Ab[(size_t)i * NX + k], lb[i], acc);
    out[(size_t)b * NX + k] = acc * qinv;
}

// ---------------------------------------------------------------------------
extern "C" void kernel_launch(void* const* d_in, const int* in_sizes, int n_in,
                              void* d_out, int out_size, void* d_ws, size_t ws_size,
                              hipStream_t stream) {
    const float* puzz = (const float*)d_in[0];   // 64*9*9*9
    const float* A    = (const float*)d_in[1];   // 64*324*729
    const float* bvec = (const float*)d_in[2];   // 64*324
    const float* Q    = (const float*)d_in[3];   // 729*729 (= q*I)
    float* out = (float*)d_out;                  // 64*729

    // workspace: S (64*336*336 f32 = 28.9 MB) then r/lam (64*336 f32)
    float* S = (float*)d_ws;
    float* r = S + (size_t)BATCH * LD * LD;

    optnet_rhs<<<dim3((NEQ + 127) / 128, BATCH), 128, 0, stream>>>(puzz, A, bvec, Q, r);
    optnet_s_tiles<<<dim3(NTB, NTB, BATCH), 32, 0, stream>>>(A, S);
    optnet_chol<<<BATCH, 256, 0, stream>>>(S);
    optnet_solve<<<BATCH, 256, 0, stream>>>(S, r);
    optnet_recover<<<dim3((NX + 255) / 256, BATCH), 256, 0, stream>>>(puzz, A, Q, r, out);
}